// DendriticBranchLayerSparse_62689342652745
// MI455X (gfx1250) — compile-verified
//
#include <hip/hip_runtime.h>

// out[b,o] = sum_{k<4} w[o*4+k] * x[b, o*4+k] + t[b] * tw[o]
// Memory-bound (0.5 FLOP/byte): stream x with NT b128 loads, weights staged in
// LDS via CDNA5 async-to-LDS, rank-1 bias fused with V_WMMA_F32_16X16X4_F32.

#define BATCH   1024
#define NUM_OUT 8192
#define BF      4
#define NUM_IN  (NUM_OUT * BF)

#define O_PER_BLOCK 128   // 8 waves * 16 outputs
#define B_TILES     4     // 4 x 16 batch rows per block
#define THREADS     256

typedef __attribute__((ext_vector_type(2))) float v2f;
typedef __attribute__((ext_vector_type(4))) float v4f;
typedef __attribute__((ext_vector_type(8))) float v8f;
typedef __attribute__((ext_vector_type(4))) int   v4i;

// pointer-to-AS1-int4 and pointer-to-AS3-int4 (async-to-LDS builtin operands)
typedef __attribute__((address_space(1))) v4i* gp4;
typedef __attribute__((address_space(3))) v4i* lp4;

__global__ __launch_bounds__(THREADS) void dendritic_wmma_kernel(
    const float* __restrict__ x,
    const float* __restrict__ t,
    const float* __restrict__ w,
    const float* __restrict__ tw,
    float* __restrict__ out)
{
    __shared__ __align__(16) float lw[O_PER_BLOCK * BF]; // 2 KB  (branch weights)
    __shared__ __align__(16) float ltw[O_PER_BLOCK];     // 512 B (t_weights)

    const int tid  = threadIdx.x;
    const int wave = tid >> 5;
    const int lane = tid & 31;
    const int n    = lane & 15;   // output column within 16-wide tile
    const int hi   = lane >> 4;   // row-half select (WMMA C/D layout)

    const int o_base = blockIdx.x * O_PER_BLOCK;
    const int o      = o_base + wave * 16 + n;

    // ---- stage this block's weights into LDS (CDNA5 async path) ----
#if __has_builtin(__builtin_amdgcn_global_load_async_to_lds_b128)
    if (tid < 128) {
        __builtin_amdgcn_global_load_async_to_lds_b128(
            (gp4)(w + (size_t)o_base * BF + tid * 4),
            (lp4)(lw + tid * 4), 0, 0);
    } else if (tid < 160) {
        const int i = tid - 128;
        __builtin_amdgcn_global_load_async_to_lds_b128(
            (gp4)(tw + (size_t)o_base + i * 4),
            (lp4)(ltw + i * 4), 0, 0);
    }
#if __has_builtin(__builtin_amdgcn_s_wait_asynccnt)
    __builtin_amdgcn_s_wait_asynccnt(0);
#else
    asm volatile("s_wait_asynccnt 0" ::: "memory");
#endif
#else
    if (tid < 128) {
        ((v4f*)lw)[tid] = ((const v4f*)(w + (size_t)o_base * BF))[tid];
    } else if (tid < 160) {
        ((v4f*)ltw)[tid - 128] = ((const v4f*)(tw + o_base))[tid - 128];
    }
#endif
    __syncthreads();

    // per-lane weights for output column o (LDS, reused over all batch tiles)
    const v4f   wv  = ((const v4f*)lw)[wave * 16 + n];
    const float twv = ltw[wave * 16 + n];

    const int b_block = blockIdx.y * (B_TILES * 16);

    for (int j = 0; j < B_TILES; ++j) {
        const int b0 = b_block + j * 16;

        // Accumulator laid out exactly as WMMA 16x16 f32 C/D:
        // acc[r] holds element (M = r + 8*hi, N = lane&15)
        v8f acc;
#pragma unroll
        for (int r = 0; r < 8; ++r) {
            const int m = b0 + r + 8 * hi;
            const v4f xv = __builtin_nontemporal_load(
                (const v4f*)(x + (size_t)m * NUM_IN + (size_t)o * BF));
            acc[r] = __builtin_fmaf(wv[3], xv[3],
                     __builtin_fmaf(wv[2], xv[2],
                     __builtin_fmaf(wv[1], xv[1], wv[0] * xv[0])));
        }

        // Rank-1 bias t[b] * tw[o] fused via full-precision f32 WMMA:
        // A (16x4): column k=0 = t[b0..b0+15];  B (4x16): row k=0 = tw tile.
        const float tval = (hi == 0) ? t[b0 + n] : 0.0f;

#if __has_builtin(__builtin_amdgcn_wmma_f32_16x16x4_f32)
        v2f a;  a[0] = tval;                       a[1] = 0.0f;
        v2f bm; bm[0] = (hi == 0) ? twv : 0.0f;    bm[1] = 0.0f;
        acc = __builtin_amdgcn_wmma_f32_16x16x4_f32(
            false, a, false, bm, (short)0, acc, false, false);
#else
#pragma unroll
        for (int r = 0; r < 8; ++r)
            acc[r] = __builtin_fmaf(t[b0 + r + 8 * hi], twv, acc[r]);
#endif

#pragma unroll
        for (int r = 0; r < 8; ++r) {
            const int m = b0 + r + 8 * hi;
            __builtin_nontemporal_store(acc[r], out + (size_t)m * NUM_OUT + o);
        }
    }
}

extern "C" void kernel_launch(void* const* d_in, const int* in_sizes, int n_in,
                              void* d_out, int out_size, void* d_ws, size_t ws_size,
                              hipStream_t stream) {
    const float* x  = (const float*)d_in[0];   // [1024, 32768]
    const float* t  = (const float*)d_in[1];   // [1024]
    const float* w  = (const float*)d_in[2];   // [32768]
    const float* tw = (const float*)d_in[3];   // [8192, 1]
    float* out = (float*)d_out;                // [1024, 8192]

    dim3 grid(NUM_OUT / O_PER_BLOCK, BATCH / (B_TILES * 16)); // (64, 16)
    dendritic_wmma_kernel<<<grid, THREADS, 0, stream>>>(x, t, w, tw, out);
}